// GATPolicy_69191923138732
// MI455X (gfx1250) — compile-verified
//
#include <hip/hip_runtime.h>
#include <hip/hip_bf16.h>
#include <stdint.h>

// ---------------------------------------------------------------------------
// GATPolicy forward for MI455X (gfx1250), fp32 end-to-end.
// All dense layers use V_WMMA_F32_16X16X4_F32 (f32 in / f32 acc) so results
// match the fp32 JAX reference; the workload (~170 GFLOP after algebraic
// simplification, ~400 MB traffic) is HBM/scatter bound at 23.3 TB/s.
// Key cuts vs the naive graph:
//   * edge_attr == 0  =>  We GEMM and att_e term eliminated entirely.
//   * concat([right,x]) @ pc_W  =>  right@Wtop + x@Wbot (accumulating GEMM).
//   * cartesian-product fc0 (32768x1536x768) factored into two tiny GEMMs
//     (pe@Wtop, u@Wbot) + broadcast add: 77 GFLOP -> 0.6 GFLOP.
// GEMM v2: B tiles are staged through LDS (double-buffered, swizzled into the
// per-lane WMMA fragment layout, conflict-free banks) so the 4 waves of a
// block share one copy of B: 4x less global B traffic, inner loop reads B
// with single ds_load_b64 per fragment.
// ---------------------------------------------------------------------------

#define D_   768
#define HD_  1536
#define NU_  256
#define NP_  8192
#define NK_  128
#define F_   3551
#define NPAIR_ (NU_ * NK_)

typedef __attribute__((ext_vector_type(2))) float v2f;
typedef __attribute__((ext_vector_type(8))) float v8f;

enum { GF_ACCUM = 1, GF_RELU = 2, GF_RELUA = 4, GF_BN = 8 };

// 1/sqrt(1 + 1e-5): eval-mode BatchNorm with running stats (0,1)
#define BN_SCALE 0.9999950000374997f

// LDS staging geometry: KSTEP=16 rows x 64 cols of B per round.
// Fragment-swizzled index for element (kl, n), kl in [0,16), n in [0,64):
//   quad q = kl>>2, row-pair p = (kl>>1)&1, row-in-pair r = kl&1
//   lidx = q*320 + p*160 + n*2 + r
// p-stride 160 (== 32 mod 64 banks) makes the compute-side ds_load_b64 of
// lanes 0-15 (p=half=0) and lanes 16-31 (p=1) land on disjoint bank halves.
#define KSTEP_  16
#define BUFSZ_  1280   // >= 3*320 + 160 + 63*2 + 1 + 1

__device__ __forceinline__ void stage_gload(float st[8], const float* __restrict__ W,
                                            int k0, int K, int N, int tileN, int t)
{
    #pragma unroll
    for (int i = 0; i < 8; ++i) {
        const int idx = t + 128 * i;          // 0..1023
        const int kl  = idx >> 6;             // 0..15
        const int n   = idx & 63;             // coalesced along N
        const int kk  = k0 + kl;
        st[i] = (kk < K) ? W[(size_t)kk * N + tileN + n] : 0.f;
    }
}

__device__ __forceinline__ void stage_sstore(float* __restrict__ buf, const float st[8], int t)
{
    #pragma unroll
    for (int i = 0; i < 8; ++i) {
        const int idx = t + 128 * i;
        const int kl  = idx >> 6;
        const int n   = idx & 63;
        buf[(kl >> 2) * 320 + ((kl >> 1) & 1) * 160 + n * 2 + (kl & 1)] = st[i];
    }
}

// ---------------------------------------------------------------------------
// Generic GEMM: C[M,N] (+)= act( bn( A[M,K] @ W[K,N] + bias ) )
// Block = 4 waves = 64(M) x 64(N) tile; one wave = 16(M) x 64(N) with 4 f32
// WMMA accumulators. Requires M % 64 == 0 (true at every call site) so all
// waves participate in staging/barriers.
// A fragment (v2f): lane<16 holds K=k..k+1 of row M=lane; lane>=16 K=k+2..k+3.
// C/D (v8f): element e -> row tileM + e + 8*(lane>>4), col tileN + (lane&15).
// ---------------------------------------------------------------------------
__global__ __launch_bounds__(128)
void gemm_wmma_f32(const float* __restrict__ A, const float* __restrict__ W,
                   const float* __restrict__ bias,
                   const float* __restrict__ bng, const float* __restrict__ bnb,
                   float* __restrict__ C, int M, int N, int K, int flags)
{
    __shared__ float lds[2 * BUFSZ_];

    const int lane  = threadIdx.x;
    const int t     = threadIdx.y * 32 + lane;      // 0..127
    const int tileM = (blockIdx.y * 4 + threadIdx.y) * 16;
    const int tileN = blockIdx.x * 64;
    const int half  = lane >> 4;                    // 0 | 1
    const int l16   = lane & 15;
    const bool reluA = (flags & GF_RELUA) != 0;

    v8f acc0 = {}, acc1 = {}, acc2 = {}, acc3 = {};
    const float* Arow = A + (size_t)(tileM + l16) * K;

    const int nkb = (K + KSTEP_ - 1) / KSTEP_;
    float st[8];

    // prologue: stage round 0 into buffer 0
    stage_gload(st, W, 0, K, N, tileN, t);
    stage_sstore(lds, st, t);
    __syncthreads();

    for (int kb = 0; kb < nkb; ++kb) {
        const int k0 = kb * KSTEP_;
        const float* cur = lds + (kb & 1) * BUFSZ_;

        // issue next round's global loads before compute (latency overlap)
        if (kb + 1 < nkb) stage_gload(st, W, (kb + 1) * KSTEP_, K, N, tileN, t);

        if (k0 + KSTEP_ <= K) {                     // full chunk: no guards
            #pragma unroll
            for (int q = 0; q < 4; ++q) {
                const int ak = k0 + q * 4 + 2 * half;
                v2f a;
                a.x = Arow[ak];
                a.y = Arow[ak + 1];
                if (reluA) { a.x = fmaxf(a.x, 0.f); a.y = fmaxf(a.y, 0.f); }
                const float* bp = cur + q * 320 + half * 160 + l16 * 2;
                v2f b0 = *(const v2f*)(bp);
                v2f b1 = *(const v2f*)(bp + 32);
                v2f b2 = *(const v2f*)(bp + 64);
                v2f b3 = *(const v2f*)(bp + 96);
                acc0 = __builtin_amdgcn_wmma_f32_16x16x4_f32(false, a, false, b0, (short)0, acc0, false, false);
                acc1 = __builtin_amdgcn_wmma_f32_16x16x4_f32(false, a, false, b1, (short)0, acc1, false, false);
                acc2 = __builtin_amdgcn_wmma_f32_16x16x4_f32(false, a, false, b2, (short)0, acc2, false, false);
                acc3 = __builtin_amdgcn_wmma_f32_16x16x4_f32(false, a, false, b3, (short)0, acc3, false, false);
            }
        } else {                                     // K tail: guarded A, B pre-zeroed
            #pragma unroll
            for (int q = 0; q < 4; ++q) {
                const int ak = k0 + q * 4 + 2 * half;
                v2f a = {0.f, 0.f};
                if (ak < K)     { float v = Arow[ak];     a.x = reluA ? fmaxf(v, 0.f) : v; }
                if (ak + 1 < K) { float v = Arow[ak + 1]; a.y = reluA ? fmaxf(v, 0.f) : v; }
                const float* bp = cur + q * 320 + half * 160 + l16 * 2;
                v2f b0 = *(const v2f*)(bp);
                v2f b1 = *(const v2f*)(bp + 32);
                v2f b2 = *(const v2f*)(bp + 64);
                v2f b3 = *(const v2f*)(bp + 96);
                acc0 = __builtin_amdgcn_wmma_f32_16x16x4_f32(false, a, false, b0, (short)0, acc0, false, false);
                acc1 = __builtin_amdgcn_wmma_f32_16x16x4_f32(false, a, false, b1, (short)0, acc1, false, false);
                acc2 = __builtin_amdgcn_wmma_f32_16x16x4_f32(false, a, false, b2, (short)0, acc2, false, false);
                acc3 = __builtin_amdgcn_wmma_f32_16x16x4_f32(false, a, false, b3, (short)0, acc3, false, false);
            }
        }

        // ping-pong: write next buffer (its last readers synced a round ago),
        // then one barrier makes it visible for round kb+1.
        if (kb + 1 < nkb) {
            stage_sstore(lds + ((kb + 1) & 1) * BUFSZ_, st, t);
            __syncthreads();
        }
    }

    // fused epilogue: accumulate / bias / eval-BN / ReLU
    #pragma unroll
    for (int e = 0; e < 8; ++e) {
        const int r = tileM + e + half * 8;
        float* Cp = C + (size_t)r * N + tileN + l16;
        float vv[4] = { acc0[e], acc1[e], acc2[e], acc3[e] };
        #pragma unroll
        for (int j = 0; j < 4; ++j) {
            const int c = tileN + j * 16 + l16;
            float v = vv[j];
            if (flags & GF_ACCUM) v += Cp[j * 16];
            if (bias)             v += bias[c];
            if (flags & GF_BN)    v = v * (bng[c] * BN_SCALE) + bnb[c];
            if (flags & GF_RELU)  v = fmaxf(v, 0.f);
            Cp[j * 16] = v;
        }
    }
}

// ---- attention logits: out[n,h] = sum_c X[n, h*768+c] * att[h*768+c] -------
__global__ void att_reduce_kernel(const float* __restrict__ X,
                                  const float* __restrict__ att,
                                  float* __restrict__ out)
{
    const int n = blockIdx.x, h = blockIdx.y, lane = threadIdx.x;
    const float* xp = X + (size_t)n * HD_ + h * D_;
    const float* ap = att + h * D_;
    float s = 0.f;
    for (int c = lane; c < D_; c += 32) s += xp[c] * ap[c];
    for (int off = 16; off; off >>= 1) s += __shfl_down(s, off, 32);
    if (lane == 0) out[n * 2 + h] = s;
}

// ---- small utility kernels -------------------------------------------------
__global__ void fill_i32_kernel(int* p, int v, int n) {
    int i = blockIdx.x * blockDim.x + threadIdx.x;
    if (i < n) p[i] = v;
}
__global__ void fill_f32_kernel(float* p, float v, int n) {
    int i = blockIdx.x * blockDim.x + threadIdx.x;
    if (i < n) p[i] = v;
}
__global__ void bias_fill_kernel(float* __restrict__ out, const float* __restrict__ bias,
                                 size_t total, int cols) {
    size_t i = (size_t)blockIdx.x * blockDim.x + threadIdx.x;
    const size_t stride = (size_t)gridDim.x * blockDim.x;
    for (; i < total; i += stride) out[i] = bias[i % cols];
}
__global__ void imin_kernel(const int* __restrict__ x, int n, int* out) {
    int lm = 0x7fffffff;
    for (int i = blockIdx.x * blockDim.x + threadIdx.x; i < n; i += gridDim.x * blockDim.x)
        lm = min(lm, x[i]);
    for (int off = 16; off; off >>= 1) lm = min(lm, __shfl_down(lm, off, 32));
    if ((threadIdx.x & 31) == 0) atomicMin(out, lm);
}

// ---- order-preserving float<->int map for atomic segment-max ---------------
__device__ __forceinline__ int f32_ord(float f) {
    int b = __float_as_int(f);
    return b >= 0 ? b : (b ^ 0x7fffffff);
}
__device__ __forceinline__ float ord_f32(int o) {
    int b = o >= 0 ? o : (o ^ 0x7fffffff);
    return __int_as_float(b);
}

// ---- edge pass 1: a = leaky_relu(as[src]+ad[dst]); segment max over dst ----
__global__ void edge_pass1_kernel(const int* __restrict__ src, const int* __restrict__ sminp,
                                  const int* __restrict__ dst, const int* __restrict__ dminp,
                                  const float* __restrict__ as_, const float* __restrict__ ad_,
                                  float* __restrict__ aout, int* __restrict__ amax, int E)
{
    int e = blockIdx.x * blockDim.x + threadIdx.x;
    if (e >= E) return;
    const int s = src[e] - *sminp;
    const int d = dst[e] - *dminp;
    #pragma unroll
    for (int h = 0; h < 2; ++h) {
        float a = as_[s * 2 + h] + ad_[d * 2 + h];
        a = a > 0.f ? a : 0.2f * a;           // leaky_relu 0.2
        aout[e * 2 + h] = a;
        atomicMax(&amax[d * 2 + h], f32_ord(a));
    }
}

// ---- edge pass 2: e = exp(a - amax[dst]); segment sum over dst -------------
__global__ void edge_pass2_kernel(const int* __restrict__ dst, const int* __restrict__ dminp,
                                  float* __restrict__ aexp, const int* __restrict__ amax,
                                  float* __restrict__ den, int E)
{
    int e = blockIdx.x * blockDim.x + threadIdx.x;
    if (e >= E) return;
    const int d = dst[e] - *dminp;
    #pragma unroll
    for (int h = 0; h < 2; ++h) {
        float m  = ord_f32(amax[d * 2 + h]);
        float ee = __expf(aexp[e * 2 + h] - m);
        aexp[e * 2 + h] = ee;
        atomicAdd(&den[d * 2 + h], ee);
    }
}

// ---- weighted scatter: g[dst, h*768+c] += xs[src, h*768+c] * alpha[e,h] ----
__global__ void scatter_agg_kernel(const int* __restrict__ src, const int* __restrict__ sminp,
                                   const int* __restrict__ dst, const int* __restrict__ dminp,
                                   const float* __restrict__ xs, const float* __restrict__ eexp,
                                   const float* __restrict__ den, float* __restrict__ g)
{
    const int e = blockIdx.x;
    const int s = src[e] - *sminp;
    const int d = dst[e] - *dminp;
    const float a0 = eexp[e * 2 + 0] / (den[d * 2 + 0] + 1e-16f);
    const float a1 = eexp[e * 2 + 1] / (den[d * 2 + 1] + 1e-16f);
    const float* xp = xs + (size_t)s * HD_;
    float*       gp = g  + (size_t)d * HD_;
    for (int j = threadIdx.x; j < HD_; j += blockDim.x) {
        const float al = (j < D_) ? a0 : a1;
        atomicAdd(&gp[j], xp[j] * al);
    }
}

// ---- fc0 via factored cartesian product + BN + ReLU ------------------------
// h0[r, :] = relu(bn0(P1[r % Nk, :] + P2[r / Nk, :] + b))
__global__ void pairs_fc0_kernel(const float* __restrict__ P1, const float* __restrict__ P2,
                                 const float* __restrict__ b, const float* __restrict__ bng,
                                 const float* __restrict__ bnb, float* __restrict__ h0)
{
    const int r = blockIdx.x;
    const float* p1 = P1 + (size_t)(r % NK_) * D_;
    const float* p2 = P2 + (size_t)(r / NK_) * D_;
    float* o = h0 + (size_t)r * D_;
    for (int j = threadIdx.x; j < D_; j += blockDim.x) {
        float v = p1[j] + p2[j] + b[j];
        v = v * (bng[j] * BN_SCALE) + bnb[j];
        o[j] = fmaxf(v, 0.f);
    }
}

// ---- fc3 matvec (K=128) + sigmoid, one wave per row ------------------------
__global__ void fc3_sigmoid_kernel(const float* __restrict__ X, const float* __restrict__ w,
                                   const float* __restrict__ b, float* __restrict__ out)
{
    const int r = blockIdx.x, lane = threadIdx.x;
    const float* xp = X + (size_t)r * 128;
    float s = 0.f;
    for (int c = lane; c < 128; c += 32) s += xp[c] * w[c];
    for (int off = 16; off; off >>= 1) s += __shfl_down(s, off, 32);
    if (lane == 0) out[r] = 1.f / (1.f + __expf(-(s + b[0])));
}

// ---------------------------------------------------------------------------
// Host-side orchestration
// ---------------------------------------------------------------------------
static void gemm(hipStream_t st, const float* A, const float* W, const float* bias,
                 const float* bng, const float* bnb, float* C,
                 int M, int N, int K, int flags)
{
    dim3 g(N / 64, M / 64), b(32, 4);   // all call sites: M % 64 == 0, N % 64 == 0
    gemm_wmma_f32<<<g, b, 0, st>>>(A, W, bias, bng, bnb, C, M, N, K, flags);
}

struct GatP {
    const float *Ws, *Wd, *atts, *attd, *bias, *ltW, *ltb, *ffW, *ffb, *pcW, *pcb;
};
// gat_p() insertion order: Ws Wd We att_s att_d att_e bias lt_W lt_b ff_W ff_b pc_W pc_b
static GatP gatp(const float* const* P, int base) {
    GatP g;
    g.Ws = P[base + 0]; g.Wd = P[base + 1];                 // P[base+2] = We (unused: edges are 0)
    g.atts = P[base + 3]; g.attd = P[base + 4];             // P[base+5] = att_e (unused)
    g.bias = P[base + 6];
    g.ltW = P[base + 7]; g.ltb = P[base + 8];
    g.ffW = P[base + 9]; g.ffb = P[base + 10];
    g.pcW = P[base + 11]; g.pcb = P[base + 12];
    return g;
}

// One full _gat_net: out = relu(concat([x_dst, ff(relu(lt(relu(conv))))]) @ pcW + pcb)
static void run_gat(hipStream_t st, const GatP& p,
                    const float* x_src, int Ns, const float* x_dst, int Nd,
                    const int* src, const int* sminp, const int* dst, const int* dminp, int E,
                    float* xs, float* xd, float* g, float* x1, float* x2,
                    float* as_, float* ad_, float* eexp, int* amax, float* den,
                    float* out)
{
    gemm(st, x_src, p.Ws, nullptr, nullptr, nullptr, xs, Ns, HD_, D_, 0);
    gemm(st, x_dst, p.Wd, nullptr, nullptr, nullptr, xd, Nd, HD_, D_, 0);
    att_reduce_kernel<<<dim3(Ns, 2), 32, 0, st>>>(xs, p.atts, as_);
    att_reduce_kernel<<<dim3(Nd, 2), 32, 0, st>>>(xd, p.attd, ad_);
    const int nd2 = Nd * 2;
    fill_i32_kernel<<<(nd2 + 255) / 256, 256, 0, st>>>(amax, (int)0x80000000, nd2);
    fill_f32_kernel<<<(nd2 + 255) / 256, 256, 0, st>>>(den, 0.f, nd2);
    edge_pass1_kernel<<<(E + 255) / 256, 256, 0, st>>>(src, sminp, dst, dminp, as_, ad_, eexp, amax, E);
    edge_pass2_kernel<<<(E + 255) / 256, 256, 0, st>>>(dst, dminp, eexp, amax, den, E);
    bias_fill_kernel<<<1024, 256, 0, st>>>(g, p.bias, (size_t)Nd * HD_, HD_);
    scatter_agg_kernel<<<E, 256, 0, st>>>(src, sminp, dst, dminp, xs, eexp, den, g);
    // x1 = relu(g) @ ltW + ltb ; x2 = relu(x1) @ ffW + ffb
    gemm(st, g,  p.ltW, p.ltb, nullptr, nullptr, x1, Nd, D_, HD_, GF_RELUA);
    gemm(st, x1, p.ffW, p.ffb, nullptr, nullptr, x2, Nd, D_, D_, GF_RELUA);
    // out = relu(x_dst @ pcW[:768] + x2 @ pcW[768:] + pcb)
    gemm(st, x_dst, p.pcW,                   nullptr, nullptr, nullptr, out, Nd, D_, D_, 0);
    gemm(st, x2,    p.pcW + (size_t)D_ * D_, p.pcb,   nullptr, nullptr, out, Nd, D_, D_, GF_ACCUM | GF_RELU);
}

extern "C" void kernel_launch(void* const* d_in, const int* in_sizes, int n_in,
                              void* d_out, int out_size, void* d_ws, size_t ws_size,
                              hipStream_t stream)
{
    (void)out_size; (void)ws_size;
    // ---- inputs (setup_inputs order) ----
    const float* user_f = (const float*)d_in[0];
    const int*   ei     = (const int*)d_in[1];
    /* d_in[2] = edge_features: overwritten with zeros in reference -> unused */
    const float* prod_f = (const float*)d_in[3];
    const float* pers_f = (const float*)d_in[4];
    const int*   pp     = (const int*)d_in[5];
    const int E  = in_sizes[1] / 2;
    const int E2 = in_sizes[5] / 2;
    const int* ei_prod = ei;            // edge_indices[0]
    const int* ei_user = ei + E;        // edge_indices[1]
    const int* pp_prod = pp;            // persona_prod_edge_ind[0]
    const int* pp_pers = pp + E2;       // persona_prod_edge_ind[1]

    // ---- params: d_in[6..] = leaves of params dict in insertion order ----
    const float* P[64];
    const int nP = n_in - 6;
    for (int i = 0; i < nP && i < 64; ++i) P[i] = (const float*)d_in[6 + i];
    const float *W_user = P[0], *b_user = P[1];
    const float *W_prod = P[2], *b_prod = P[3];
    const float *W_pers = P[4], *b_pers = P[5];
    GatP c2v = gatp(P, 6), v2c = gatp(P, 19), p2p = gatp(P, 32);
    const float *fc0W = P[45], *fc0b = P[46], *bn0g = P[47], *bn0b = P[48];
    const float *fc1W = P[49], *fc1b = P[50], *bn1g = P[51], *bn1b = P[52];
    const float *fc2W = P[53], *fc2b = P[54], *bn2g = P[55], *bn2b = P[56];
    const float *fc3W = P[57], *fc3b = P[58];

    // ---- workspace carve-up (floats); ~180 MB with aliasing ----
    float* wsf = (float*)d_ws;
    size_t off = 0;
    auto grab = [&](size_t n) { float* p = wsf + off; off += (n + 63) & ~(size_t)63; return p; };
    float* xs  = grab((size_t)NP_ * HD_);     // h0 spans [xs, xd] in the MLP phase
    float* xd  = grab((size_t)NP_ * HD_);     // also aliased as g (xd is dead after ad_ reduce)
    float* pr0 = grab((size_t)NP_ * D_);      // h1 spans [pr0, x1] in the MLP phase
    float* x1  = grab((size_t)NP_ * D_);
    float* x2  = grab((size_t)NP_ * D_);      // also h2
    float* u0  = grab((size_t)NU_ * D_);
    float* pe0 = grab((size_t)NK_ * D_);
    float* pe1 = grab((size_t)NK_ * D_);
    float* Pb  = grab((size_t)(NK_ + NU_) * D_);  // P1 | P2
    float* as_ = grab((size_t)NP_ * 2);
    float* ad_ = grab((size_t)NP_ * 2);
    float* eex = grab((size_t)E * 2);
    float* den = grab((size_t)NP_ * 2);
    int*  amax = (int*)grab((size_t)NP_ * 2);
    int*  icon = (int*)grab(16);              // [0]=zero, [1]=min(ei_prod), [2]=min(pp_pers)
    float* gbuf = xd;                          // alias
    float* P1 = Pb, *P2 = Pb + (size_t)NK_ * D_;

    // ---- outputs: (x[32768,1], u[256,768], pr[8192,768]) flat ----
    float* outX  = (float*)d_out;
    float* outU  = outX + NPAIR_;
    float* outPr = outU + (size_t)NU_ * D_;

    // ---- index minima (reference shifts ei0 and pp1 by their minima) ----
    fill_i32_kernel<<<1, 32, 0, stream>>>(icon, 0, 1);
    fill_i32_kernel<<<1, 32, 0, stream>>>(icon + 1, 0x7fffffff, 2);
    imin_kernel<<<64, 256, 0, stream>>>(ei_prod, E, icon + 1);
    imin_kernel<<<32, 256, 0, stream>>>(pp_pers, E2, icon + 2);

    // ---- feature projections (K = 3551, tail handled in GEMM) ----
    gemm(stream, user_f, W_user, b_user, nullptr, nullptr, u0,  NU_, D_, F_, 0);
    gemm(stream, prod_f, W_prod, b_prod, nullptr, nullptr, pr0, NP_, D_, F_, 0);
    gemm(stream, pers_f, W_pers, b_pers, nullptr, nullptr, pe0, NK_, D_, F_, 0);

    // ---- conv_c_to_v: src=user (ei1), dst=product (ei0 - min) -> pr1 ----
    run_gat(stream, c2v, u0, NU_, pr0, NP_, ei_user, icon, ei_prod, icon + 1, E,
            xs, xd, gbuf, x1, x2, as_, ad_, eex, amax, den, outPr);

    // ---- conv_v_to_c: src=product (ei0 - min), dst=user (ei1) -> u1 ----
    run_gat(stream, v2c, outPr, NP_, u0, NU_, ei_prod, icon + 1, ei_user, icon, E,
            xs, xd, gbuf, x1, x2, as_, ad_, eex, amax, den, outU);

    // ---- conv_p2p: src=product (pp0), dst=persona (pp1 - min) -> pe1 ----
    run_gat(stream, p2p, outPr, NP_, pe0, NK_, pp_prod, icon, pp_pers, icon + 2, E2,
            xs, xd, gbuf, x1, x2, as_, ad_, eex, amax, den, pe1);

    // ---- MLP head: factored fc0, fused BN+ReLU fc1/fc2, fc3+sigmoid ----
    gemm(stream, pe1,  fc0W,                   nullptr, nullptr, nullptr, P1, NK_, D_, D_, 0);
    gemm(stream, outU, fc0W + (size_t)D_ * D_, nullptr, nullptr, nullptr, P2, NU_, D_, D_, 0);
    float* h0 = xs;    // 32768 x 768 over [xs, xd] (both dead now)
    pairs_fc0_kernel<<<NPAIR_, 256, 0, stream>>>(P1, P2, fc0b, bn0g, bn0b, h0);
    float* h1 = pr0;   // 32768 x 384 over [pr0, x1]
    gemm(stream, h0, fc1W, fc1b, bn1g, bn1b, h1, NPAIR_, 384, D_, GF_BN | GF_RELU);
    float* h2 = x2;    // 32768 x 128
    gemm(stream, h1, fc2W, fc2b, bn2g, bn2b, h2, NPAIR_, 128, 384, GF_BN | GF_RELU);
    fc3_sigmoid_kernel<<<NPAIR_, 32, 0, stream>>>(h2, fc3W, fc3b, outX);
}